// EvictableKVCache_62380105007447
// MI455X (gfx1250) — compile-verified
//
#include <hip/hip_runtime.h>

#define N_ENTRIES 65536
#define VALUE_DIM 8
#define N_BITS    31
#define BATCH     1048576

#define THREADS   1024   // 32 wave32 waves = one full WGP
#define GRID      512    // 512*1024*2 = BATCH (grid-stride, 2 queries/thread)

typedef __attribute__((address_space(1))) int global_int;
typedef __attribute__((address_space(3))) int lds_int;
typedef float vfloat4 __attribute__((ext_vector_type(4)));   // builtin-compatible 16B vector

__global__ __launch_bounds__(THREADS)
void kv_lookup_kernel(const float* __restrict__ query,
                      const float* __restrict__ values,
                      const unsigned long long* __restrict__ addrs64,
                      float* __restrict__ out)
{
    // 65536 x u32 = 256 KB — needs CDNA5's 320KB-per-WGP LDS.
    extern __shared__ unsigned int lds_tab[];

    // ---- Phase 1: stage low dword of each sorted int64 key into LDS ----
    // Keys are < 2^31, so u32 order == int64 order and equality is exact.
    for (unsigned i = threadIdx.x; i < N_ENTRIES; i += THREADS) {
#if __has_builtin(__builtin_amdgcn_global_load_async_to_lds_b32)
        // Async DMA: 4 bytes per lane, global -> LDS, no VGPR round-trip.
        void* gsrc = (void*)(addrs64 + i);   // strip const; low dword of int64
        void* ldst = (void*)(lds_tab + i);
        __builtin_amdgcn_global_load_async_to_lds_b32(
            (global_int*)gsrc,
            (lds_int*)ldst,
            /*offset=*/0, /*cpol=*/0);
#else
        lds_tab[i] = (unsigned)addrs64[i];   // fallback: load + ds_store
#endif
    }
#if __has_builtin(__builtin_amdgcn_global_load_async_to_lds_b32)
#if __has_builtin(__builtin_amdgcn_s_wait_asynccnt)
    __builtin_amdgcn_s_wait_asynccnt(0);
#else
    asm volatile("s_wait_asynccnt 0x0" ::: "memory");
#endif
#endif
    __syncthreads();

    // ---- Phase 2: decode bits, LDS binary search, gather values ----
    const unsigned stride = GRID * THREADS;
    for (unsigned q = blockIdx.x * THREADS + threadIdx.x; q < BATCH; q += stride) {
        const float* qp = query + (size_t)q * N_BITS;
        if (q + stride < BATCH)   // prefetch next query row -> global_prefetch_b8
            __builtin_prefetch(qp + (size_t)stride * N_BITS, 0, 0);

        unsigned a = 0;
#pragma unroll
        for (int b = 0; b < N_BITS; ++b)
            a |= (qp[b] > 0.5f) ? (1u << b) : 0u;

        // Branchless lower bound over 2^16 sorted u32 keys in LDS:
        // lo = min(#elements < a, 65535)  (matches searchsorted + clip)
        unsigned lo = 0;
#pragma unroll
        for (int s = 15; s >= 0; --s) {
            unsigned mid = lo + (1u << s);
            lo = (lds_tab[mid - 1] < a) ? mid : lo;
        }
        const unsigned idx = lo;                   // always <= 65535
        const bool hit = (lds_tab[idx] == a);

        // Unconditional L2 gather (idx always valid), select on hit.
        const vfloat4* vp = (const vfloat4*)(values + (size_t)idx * VALUE_DIM);
        vfloat4 v0 = vp[0];
        vfloat4 v1 = vp[1];
        const vfloat4 z = (vfloat4)(0.0f);
        vfloat4 o0 = hit ? v0 : z;
        vfloat4 o1 = hit ? v1 : z;

        // Write-once 32MB output stream: non-temporal stores keep L2 free
        // for the hot key/value working set.
        vfloat4* op = (vfloat4*)(out + (size_t)q * VALUE_DIM);
        __builtin_nontemporal_store(o0, op);
        __builtin_nontemporal_store(o1, op + 1);
    }
}

extern "C" void kernel_launch(void* const* d_in, const int* in_sizes, int n_in,
                              void* d_out, int out_size, void* d_ws, size_t ws_size,
                              hipStream_t stream) {
    (void)in_sizes; (void)n_in; (void)out_size; (void)d_ws; (void)ws_size;

    const float* query                  = (const float*)d_in[0];
    const float* values                 = (const float*)d_in[1];
    const unsigned long long* addrs64   = (const unsigned long long*)d_in[2];
    float* out                          = (float*)d_out;

    const size_t shmem = (size_t)N_ENTRIES * sizeof(unsigned);  // 256 KB dynamic LDS

    // Opt-in to >default dynamic LDS. Deterministic + idempotent: called
    // unconditionally every launch (no static guards).
    (void)hipFuncSetAttribute((const void*)kv_lookup_kernel,
                              hipFuncAttributeMaxDynamicSharedMemorySize,
                              (int)shmem);

    kv_lookup_kernel<<<GRID, THREADS, shmem, stream>>>(query, values, addrs64, out);
}